// ClusterLoss_48155173322752
// MI455X (gfx1250) — compile-verified
//
#include <hip/hip_runtime.h>

typedef float v2f __attribute__((ext_vector_type(2)));
typedef float v4f __attribute__((ext_vector_type(4)));
typedef float v8f __attribute__((ext_vector_type(8)));

// Problem constants (from reference: T=4096, M=64, K=256)
constexpr int T_DIM = 4096;
constexpr int M_DIM = 64;
constexpr int K_DIM = 256;
constexpr int MK    = M_DIM * K_DIM;          // 16384
constexpr int CH    = 32;                     // chunks over T
constexpr int ROWS_PER_BLOCK = T_DIM / CH;    // 128
constexpr int WAVES = 8;                      // 256 threads
constexpr int ROWS_PER_WAVE = ROWS_PER_BLOCK / WAVES; // 16
constexpr int WS_FLOATS = MK + 1;             // colsum + entropy accumulator

__global__ void zero_ws_kernel(float* __restrict__ p, int n) {
  int i = blockIdx.x * 256 + threadIdx.x;
  if (i < n) p[i] = 0.0f;
}

__device__ __forceinline__ void accum4(const v4f a, const float mx,
                                       float& s, float& w) {
#pragma unroll
  for (int j = 0; j < 4; ++j) {
    float t = a[j] - mx;
    float e = __expf(t);
    s += e;
    w += t * e;
  }
}

// One block handles block-column m, rows [c*128, c*128+128).
// Streams 128 KB with NT b128 loads; emits per-row entropies and
// per-column partial sums.
__global__ void __launch_bounds__(256)
entropy_main_kernel(const float* __restrict__ x,
                    float* __restrict__ g_colsum,
                    float* __restrict__ g_entsum) {
  __shared__ float lds_col[K_DIM];
  __shared__ float lds_ent;

  const int tid  = threadIdx.x;
  const int lane = tid & 31;
  const int wave = tid >> 5;
  const int m    = blockIdx.x & (M_DIM - 1);
  const int c    = blockIdx.x >> 6;

  if (tid < K_DIM) lds_col[tid] = 0.0f;
  if (tid == 0) lds_ent = 0.0f;
  __syncthreads();

  const int row0 = c * ROWS_PER_BLOCK + wave * ROWS_PER_WAVE;
  const size_t col_base = (size_t)m * K_DIM;

  v4f cs0 = {0.f, 0.f, 0.f, 0.f};
  v4f cs1 = {0.f, 0.f, 0.f, 0.f};
  float ent_acc = 0.0f;

  for (int r = 0; r < ROWS_PER_WAVE; ++r) {
    const size_t rowbase = (size_t)(row0 + r) * MK + col_base;
    // lane covers columns [lane*4, lane*4+4) and [128+lane*4, 128+lane*4+4)
    const v4f a0 = __builtin_nontemporal_load((const v4f*)(x + rowbase) + lane);
    const v4f a1 = __builtin_nontemporal_load((const v4f*)(x + rowbase + 128) + lane);

    float mx = fmaxf(fmaxf(fmaxf(a0.x, a0.y), fmaxf(a0.z, a0.w)),
                     fmaxf(fmaxf(a1.x, a1.y), fmaxf(a1.z, a1.w)));
#pragma unroll
    for (int d = 16; d >= 1; d >>= 1) mx = fmaxf(mx, __shfl_xor(mx, d, 32));

    float s = 0.0f, w = 0.0f;
    accum4(a0, mx, s, w);
    accum4(a1, mx, s, w);
#pragma unroll
    for (int d = 16; d >= 1; d >>= 1) {
      s += __shfl_xor(s, d, 32);
      w += __shfl_xor(w, d, 32);
    }
    // entropy = log(sum e) - sum((x-m)e)/sum e  (uniform across lanes)
    ent_acc += __logf(s) - w / s;

    cs0 += a0;
    cs1 += a1;
  }

  // cross-wave combine in LDS (ds_add_f32)
#pragma unroll
  for (int j = 0; j < 4; ++j) {
    atomicAdd(&lds_col[lane * 4 + j], cs0[j]);
    atomicAdd(&lds_col[128 + lane * 4 + j], cs1[j]);
  }
  if (lane == 0) atomicAdd(&lds_ent, ent_acc);
  __syncthreads();

  // one global f32 atomic per column per block
  if (tid < K_DIM) {
    __hip_atomic_fetch_add(&g_colsum[col_base + tid], lds_col[tid],
                           __ATOMIC_RELAXED, __HIP_MEMORY_SCOPE_AGENT);
  }
  if (tid == 0) {
    __hip_atomic_fetch_add(g_entsum, lds_ent,
                           __ATOMIC_RELAXED, __HIP_MEMORY_SCOPE_AGENT);
  }
}

// Finalize: entropy of the 64 rows of colsum/T, combined with the L1 sum.
// Row sums of exp / (x-m)*exp are computed with V_WMMA_F32_16X16X4_F32
// (A = 16x4 data fragment, B = ones -> D accumulates per-row sums).
__global__ void __launch_bounds__(128)
finalize_kernel(const float* __restrict__ g_colsum,
                const float* __restrict__ g_entsum,
                float* __restrict__ out) {
  __shared__ float mx_s[M_DIM];
  __shared__ float s2_sum;

  const int tid  = threadIdx.x;
  const int lane = tid & 31;
  const int wave = tid >> 5;   // 4 waves, each owns 16 rows
  const float invT = 1.0f / (float)T_DIM;

  if (tid == 0) s2_sum = 0.0f;
  __syncthreads();

  // Phase 1: per-row max of (colsum * invT); wave w handles rows w*16..+15
  for (int r = 0; r < 16; ++r) {
    const int row = wave * 16 + r;
    const float* p = g_colsum + (size_t)row * K_DIM;
    const v4f a0 = *((const v4f*)p + lane);
    const v4f a1 = *((const v4f*)(p + 128) + lane);
    float mx = fmaxf(fmaxf(fmaxf(a0.x, a0.y), fmaxf(a0.z, a0.w)),
                     fmaxf(fmaxf(a1.x, a1.y), fmaxf(a1.z, a1.w)));
#pragma unroll
    for (int d = 16; d >= 1; d >>= 1) mx = fmaxf(mx, __shfl_xor(mx, d, 32));
    if (lane == 0) mx_s[row] = mx * invT;
  }
  __syncthreads();

  // Phase 2: WMMA row-sum chains for this wave's 16-row group.
  // ISA 32-bit A 16x4 layout: lanes 0-15 hold rows, {K=0,K=1}; lanes 16-31 {K=2,K=3}.
  const int rbase = wave * 16;
  const int rrow  = rbase + (lane & 15);
  const int koff  = (lane >> 4) * 2;
  const float* rowp = g_colsum + (size_t)rrow * K_DIM;
  const float rmx = mx_s[rrow];

  v8f acc_s = {0.f, 0.f, 0.f, 0.f, 0.f, 0.f, 0.f, 0.f};
  v8f acc_w = {0.f, 0.f, 0.f, 0.f, 0.f, 0.f, 0.f, 0.f};
  const v2f bones = {1.0f, 1.0f};

  for (int k0 = 0; k0 < K_DIM; k0 += 4) {
    float x0 = rowp[k0 + koff] * invT - rmx;
    float x1 = rowp[k0 + koff + 1] * invT - rmx;
    float e0 = __expf(x0);
    float e1 = __expf(x1);
    v2f ae = {e0, e1};
    v2f aw = {x0 * e0, x1 * e1};
    acc_s = __builtin_amdgcn_wmma_f32_16x16x4_f32(
        false, ae, false, bones, (short)0, acc_s, false, false);
    acc_w = __builtin_amdgcn_wmma_f32_16x16x4_f32(
        false, aw, false, bones, (short)0, acc_w, false, false);
  }

  // D layout: VGPR v, lanes 0-15 -> row v (col=lane); lanes 16-31 -> row v+8.
  // Column 0 lives in lane 0 and lane 16.
  if (lane == 0 || lane == 16) {
    const int radd = (lane == 16) ? 8 : 0;
    float part = 0.0f;
#pragma unroll
    for (int v = 0; v < 8; ++v) {
      float s = acc_s[v];
      float w = acc_w[v];
      part += __logf(s) - w / s;   // entropy of row rbase+radd+v
    }
    atomicAdd(&s2_sum, part);
    (void)radd;
  }
  __syncthreads();

  if (tid == 0) {
    const float S1 = *g_entsum;                       // sum of T*M entropies
    // result = mean(ent) - mean(ent(batch-mean))  (LMBDA = 1)
    out[0] = S1 / (float)(T_DIM * M_DIM) - s2_sum / (float)M_DIM;
  }
}

extern "C" void kernel_launch(void* const* d_in, const int* in_sizes, int n_in,
                              void* d_out, int out_size, void* d_ws, size_t ws_size,
                              hipStream_t stream) {
  const float* x = (const float*)d_in[0];   // [T, M*K] f32
  // d_in[1] is M (compile-time constant here)
  float* ws  = (float*)d_ws;                // [0,16384): colsum, [16384]: entropy sum
  float* out = (float*)d_out;

  zero_ws_kernel<<<(WS_FLOATS + 255) / 256, 256, 0, stream>>>(ws, WS_FLOATS);
  entropy_main_kernel<<<M_DIM * CH, 256, 0, stream>>>(x, ws, ws + MK);
  finalize_kernel<<<1, 128, 0, stream>>>(ws, ws + MK, out);
}